// _ESRNN_84542136255255
// MI455X (gfx1250) — compile-verified
//
#include <hip/hip_runtime.h>
#include <stdint.h>

#define N_SERIES 2048
#define N_TIME   1024
#define IN_SZ    28
#define OUT_SZ   7
#define M_SEAS   7
#define N_S      4

typedef __attribute__((ext_vector_type(4))) unsigned int u32x4;
typedef __attribute__((ext_vector_type(8))) unsigned int u32x8;

// ---------------------------------------------------------------------------
// Tensor Data Mover: 2-D tile load (global -> LDS), D# built per CDNA5 ISA §8.
//   group0: count=1 | lds_addr | global_addr[56:0] | type=2
//   group1: data_size=4B, pad ctl, tensor_dim0/1, tile_dim0/1, dim0 stride
// Issued wave-uniformly (TDM ignores EXEC); tracked by TENSORcnt.
// ---------------------------------------------------------------------------
__device__ __forceinline__ void tdm_load_2d(uint32_t lds_addr, const void* gptr,
                                            uint32_t tdim0, uint32_t tdim1,
                                            uint32_t tile0, uint32_t tile1,
                                            uint32_t stride0_elems,
                                            uint32_t pad_int_code,
                                            uint32_t pad_amt_code,
                                            uint32_t pad_en)
{
    const uint64_t ga = (uint64_t)(uintptr_t)gptr;
    u32x4 g0;
    g0.x = 1u;                                                // count=1 (valid user D#)
    g0.y = lds_addr;                                          // LDS byte address
    g0.z = (uint32_t)ga;                                      // global_addr[31:0]
    g0.w = ((uint32_t)(ga >> 32) & 0x01FFFFFFu) | (2u << 30); // ga[56:32] | type=2

    u32x8 g1;
    g1.s0 = (2u << 16) | (pad_en << 20) | (pad_int_code << 22) | (pad_amt_code << 25);
    g1.s1 = (tdim0 & 0xFFFFu) << 16;                          // tensor_dim0[15:0]
    g1.s2 = (tdim0 >> 16) | ((tdim1 & 0xFFFFu) << 16);        // tdim0[31:16]|tdim1[15:0]
    g1.s3 = (tdim1 >> 16) | ((tile0 & 0xFFFFu) << 16);        // tdim1[31:16]|tile_dim0
    g1.s4 = (tile1 & 0xFFFFu);                                // tile_dim1 (tile_dim2=0)
    g1.s5 = stride0_elems;                                    // tensor_dim0_stride[31:0]
    g1.s6 = 0u;                                               // stride0[47:32]|stride1 lo
    g1.s7 = 0u;                                               // stride1 hi
    asm volatile("tensor_load_to_lds %0, %1" :: "s"(g0), "s"(g1) : "memory");
}

// ---------------------------------------------------------------------------
// Kernel 1: per-series Holt-Winters recurrence. One series per lane.
// y tiles staged via double-buffered TDM with LDS padding (33-dword pitch ->
// bank-conflict-free column reads). Seasonal queue in registers; critical
// path per step = 1 FMA (seasonal feedback has 7-step slack).
// ---------------------------------------------------------------------------
#define K1_BS  128            // series per block (one per thread, 4 waves)
#define K1_CT  32             // timesteps per chunk
#define K1_RP  (K1_CT + 1)    // padded LDS row pitch (dwords)
#define K1_NCH (N_TIME / K1_CT)

__global__ __launch_bounds__(K1_BS)
void esrnn_recurrence(const float* __restrict__ y,
                      const float* __restrict__ embeds,
                      const int*   __restrict__ idxs,
                      float* __restrict__ levels,
                      float* __restrict__ seas)
{
    __shared__ float smem[2 * K1_BS * K1_RP];

    const int  tid    = threadIdx.x;
    const int  b0     = blockIdx.x * K1_BS;
    const int  b      = b0 + tid;
    const bool issuer = (tid < 32);                         // wave 0 drives the TDM
    const uint32_t lds_base  = (uint32_t)(uintptr_t)(&smem[0]);
    const uint32_t buf_bytes = (uint32_t)(K1_BS * K1_RP * 4);

    // per-series smoothing params
    const int   row = idxs[b];
    const float e0  = embeds[row * (2 + M_SEAS) + 0];
    const float e1  = embeds[row * (2 + M_SEAS) + 1];
    const float la  = 1.0f / (1.0f + __expf(-e0));          // lev_sms
    const float ga  = 1.0f / (1.0f + __expf(-e1));          // seas_sms
    float is[M_SEAS];
    #pragma unroll
    for (int j = 0; j < M_SEAS; ++j)
        is[j] = __expf(embeds[row * (2 + M_SEAS) + 2 + j]);

    float* seas_row = seas   + (size_t)b * (N_TIME + M_SEAS);
    float* lev_row  = levels + (size_t)b * N_TIME;
    #pragma unroll
    for (int j = 0; j < M_SEAS; ++j) seas_row[j] = is[j];
    seas_row[M_SEAS] = is[0];

    // seasonal register queue: s-values for steps t=1.. are seas[1..7] = is[1..6], is[0]
    float q0 = is[1], q1 = is[2], q2 = is[3], q3 = is[4], q4 = is[5], q5 = is[6], q6 = is[0];
    float lev = 0.0f;

    if (issuer) {                                           // prefetch chunk 0
        tdm_load_2d(lds_base, y + (size_t)b0 * N_TIME,
                    N_TIME, K1_BS, K1_CT, K1_BS, N_TIME,
                    4u /*pad every 32 dwords*/, 0u /*1 dword*/, 1u);
    }

    for (int k = 0; k < K1_NCH; ++k) {
        const int buf = k & 1;
        if (issuer) __builtin_amdgcn_s_wait_tensorcnt(0);   // chunk k landed
        __syncthreads();                                    // publish LDS to all waves
        if (issuer && (k + 1 < K1_NCH)) {                   // prefetch chunk k+1
            const int t0n = (k + 1) * K1_CT;
            tdm_load_2d(lds_base + (uint32_t)(buf ^ 1) * buf_bytes,
                        y + (size_t)b0 * N_TIME + t0n,
                        (uint32_t)(N_TIME - t0n), K1_BS, K1_CT, K1_BS, N_TIME,
                        4u, 0u, 1u);
        }

        const float* yrow = &smem[buf * K1_BS * K1_RP + tid * K1_RP];
        const int    t0   = k * K1_CT;
        #pragma unroll
        for (int t = 0; t < K1_CT; ++t) {
            const int   tt = t0 + t;
            const float yt = yrow[t];
            if (tt == 0) {
                lev = __fdividef(yt, is[0]);
                lev_row[0] = lev;
            } else {
                const float s  = q0;
                const float r  = __fdividef(yt, s);          // independent of lev chain
                const float nl = la * r + (1.0f - la) * lev; // 1-FMA critical path
                const float rs = __fdividef(yt, nl);         // 7-step slack to consume
                const float ns = ga * rs + (1.0f - ga) * s;
                lev = nl;
                lev_row[tt]           = nl;
                seas_row[tt + M_SEAS] = ns;
                q0 = q1; q1 = q2; q2 = q3; q3 = q4; q4 = q5; q5 = q6; q6 = ns;
            }
        }
        __syncthreads();                                    // reads done before buf reuse
    }
}

// ---------------------------------------------------------------------------
// Kernel 2: insample windows (dominant: ~260 MB of stores).
// One wave32 per 4 (w,b) rows: lane = (row-in-group, i0=4*(lane&7)); each lane
// produces a float4 and issues one aligned global_store_b128 (512 B/wave).
// Lanes with i0==28 copy the s_matrix tail as one aligned float4 load.
// All read sources are L2-resident; kernel is store-bandwidth bound.
// ---------------------------------------------------------------------------
__global__ __launch_bounds__(256)
void esrnn_windows_in(const float* __restrict__ y,
                      const float* __restrict__ s_matrix,
                      const float* __restrict__ levels,
                      const float* __restrict__ seas,
                      const int*   __restrict__ step_ptr,
                      float* __restrict__ out_in,
                      int n_groups)                          // = W*N_SERIES/4
{
    const int step = step_ptr[0];
    const int lane = threadIdx.x & 31;
    const int g    = blockIdx.x * 8 + (threadIdx.x >> 5);
    if (g >= n_groups) return;

    const int rr = lane >> 3;                // row within group (0..3)
    const int i0 = (lane & 7) << 2;          // element start: 0,4,...,28
    const int r  = (g << 2) + rr;            // global row index = w*2048 + b
    const int b  = r & (N_SERIES - 1);
    const int w  = r >> 11;
    const int start = w * step;

    float4 v;
    if (i0 < IN_SZ) {
        const float  lev = levels[(size_t)b * N_TIME + start + IN_SZ - 1];
        const float* yp  = y    + (size_t)b * N_TIME + start + i0;
        const float* sp  = seas + (size_t)b * (N_TIME + M_SEAS) + start + i0;
        float tmp[4];
        #pragma unroll
        for (int k = 0; k < 4; ++k)
            tmp[k] = __logf(__fdividef(yp[k], lev * sp[k]));
        v = make_float4(tmp[0], tmp[1], tmp[2], tmp[3]);
    } else {
        v = *(const float4*)(s_matrix + b * N_S);            // 16B-aligned
    }
    *(float4*)(out_in + (size_t)r * (IN_SZ + N_S) + i0) = v; // 16B-aligned b128 store
}

// ---------------------------------------------------------------------------
// Kernel 3: outsample windows. One wave handles 4 rows x 7 outputs on lanes
// 0..27; store address collapses to the contiguous out2[28*g + lane].
// ---------------------------------------------------------------------------
__global__ __launch_bounds__(256)
void esrnn_windows_out(const float* __restrict__ y,
                       const int*   __restrict__ step_ptr,
                       float* __restrict__ out_out,
                       int n_groups)                         // = W*N_SERIES/4
{
    const int step = step_ptr[0];
    const int lane = threadIdx.x & 31;
    const int g    = blockIdx.x * 8 + (threadIdx.x >> 5);
    if (g >= n_groups || lane >= 4 * OUT_SZ) return;

    const int rr = lane / OUT_SZ;            // const-7 division: mul-shift
    const int j  = lane - rr * OUT_SZ;
    const int r  = (g << 2) + rr;
    const int b  = r & (N_SERIES - 1);
    const int w  = r >> 11;
    out_out[(size_t)g * (4 * OUT_SZ) + lane] =
        y[(size_t)b * N_TIME + (size_t)w * step + IN_SZ + j];
}

// ---------------------------------------------------------------------------
extern "C" void kernel_launch(void* const* d_in, const int* in_sizes, int n_in,
                              void* d_out, int out_size, void* d_ws, size_t ws_size,
                              hipStream_t stream)
{
    (void)in_sizes; (void)n_in; (void)d_ws; (void)ws_size;
    const float* y        = (const float*)d_in[0];
    const float* s_matrix = (const float*)d_in[1];
    const float* embeds   = (const float*)d_in[2];
    const int*   idxs     = (const int*)d_in[3];
    const int*   step     = (const int*)d_in[4];
    float*       out      = (float*)d_out;

    // Recover W from out_size (deterministic, capture-safe):
    // out_size = W*B*(32+7) + B*T + B*(T+M)
    const long long fixed = (long long)N_SERIES * (2LL * N_TIME + M_SEAS);
    const int W = (int)(((long long)out_size - fixed) /
                        ((long long)N_SERIES * (IN_SZ + N_S + OUT_SZ)));

    float* out_in  = out;
    float* out_out = out_in  + (size_t)W * N_SERIES * (IN_SZ + N_S);
    float* levels  = out_out + (size_t)W * N_SERIES * OUT_SZ;
    float* seas    = levels  + (size_t)N_SERIES * N_TIME;

    esrnn_recurrence<<<N_SERIES / K1_BS, K1_BS, 0, stream>>>(y, embeds, idxs, levels, seas);

    const int n_groups = W * (N_SERIES / 4);                 // rows are 4-divisible
    const unsigned blocks = (unsigned)((n_groups + 7) / 8);
    esrnn_windows_in<<<blocks, 256, 0, stream>>>(
        y, s_matrix, levels, seas, step, out_in, n_groups);
    esrnn_windows_out<<<blocks, 256, 0, stream>>>(
        y, step, out_out, n_groups);
}